// ICEWSHAN_45646912422132
// MI455X (gfx1250) — compile-verified
//
#include <hip/hip_runtime.h>
#include <hip/hip_bf16.h>
#include <math.h>

// ============================================================================
// HAN 2-layer link prediction for MI455X (gfx1250), FP32 WMMA path.
//
// - Dense GEMMs use V_WMMA_F32_16X16X4_F32 (reference-exact fp32; pipeline is
//   scatter/bandwidth bound so the fp32 matrix ceiling is not the limiter).
// - Weight tile (128x128 = 64KB) is staged into LDS once per 8-wave block via
//   the Tensor Data Mover (tensor_load_to_lds + s_wait_tensorcnt), so each
//   A-fragment global load feeds 8 WMMAs reading B fragments from LDS.
// - Semantic attention over the 8 edge types uses an online softmax across
//   relations so per-relation outputs are never all materialized.
// ============================================================================

typedef float v2f __attribute__((ext_vector_type(2)));
typedef float v8f __attribute__((ext_vector_type(8)));
typedef unsigned int u32x4 __attribute__((ext_vector_type(4)));
typedef int i32x4 __attribute__((ext_vector_type(4)));
typedef int i32x8 __attribute__((ext_vector_type(8)));

#define D128 128
#define NSLOPE 0.2f

#if defined(__has_builtin)
#if __has_builtin(__builtin_amdgcn_tensor_load_to_lds)
#define HAVE_TDM 1
#endif
#endif

#if defined(__has_include)
#if __has_include(<hip/amd_detail/amd_gfx1250_TDM.h>)
#define TDM_6ARG 1
#endif
#endif

// ---------------------------------------------------------------------------
// TDM: DMA a 128x128 f32 tile (64KB) from global into LDS.
// D# per CDNA5 ISA 8.3/8.4: group0 = {count=1, lds_addr, global_addr, type=2},
// group1 = {data_size=4B, tensor_dim0/1=128, tile_dim0/1=128, stride0=128}.
// ---------------------------------------------------------------------------
#if HAVE_TDM
__device__ __forceinline__ void tdm_load_128x128(const float* gsrc,
                                                 void* lds_dst) {
  unsigned long long ga = (unsigned long long)(uintptr_t)gsrc;
  unsigned lds_off = (unsigned)(uintptr_t)lds_dst;  // low 32 bits = LDS offset
  u32x4 g0;
  g0[0] = 1u;                                   // count=1, user descriptor
  g0[1] = lds_off;                              // lds_addr [63:32]
  g0[2] = (unsigned)(ga & 0xFFFFFFFFull);       // global_addr lo
  g0[3] = (unsigned)((ga >> 32) & 0x01FFFFFFull) | (2u << 30);  // addr hi|type=2
  i32x8 g1;
  g1[0] = (2 << 16);      // data_size = 2 (4 bytes); no multicast/pad/iterate
  g1[1] = (128 << 16);    // tensor_dim0[15:0]  @ bits 63:48
  g1[2] = (128 << 16);    // tensor_dim1[15:0]  @ bits 95:80
  g1[3] = (128 << 16);    // tile_dim0          @ bits 127:112
  g1[4] = 128;            // tile_dim1          @ bits 143:128
  g1[5] = 128;            // tensor_dim0_stride lo32
  g1[6] = 0;
  g1[7] = 0;
  i32x4 g2 = {0, 0, 0, 0};
  i32x4 g3 = {0, 0, 0, 0};
#if TDM_6ARG
  i32x8 g4 = {0, 0, 0, 0, 0, 0, 0, 0};
  __builtin_amdgcn_tensor_load_to_lds(g0, g1, g2, g3, g4, 0);
#else
  __builtin_amdgcn_tensor_load_to_lds(g0, g1, g2, g3, 0);
#endif
}
#endif

// stage B (128x128 f32) into LDS: TDM if available, cooperative copy otherwise
__device__ __forceinline__ void stage_B_to_lds(const float* B, float* Bs) {
#if HAVE_TDM
  if (threadIdx.x == 0) {
    tdm_load_128x128(B, Bs);
    __builtin_amdgcn_s_wait_tensorcnt((short)0);
  }
#else
  for (int i = threadIdx.x; i < (D128 * D128) / 4; i += (int)blockDim.x)
    ((float4*)Bs)[i] = ((const float4*)B)[i];
#endif
  __syncthreads();
}

// ---------------------------------------------------------------------------
// utility fills
// ---------------------------------------------------------------------------
__global__ void k_fill_f32(float* __restrict__ p, float v, long n) {
  long i = (long)blockIdx.x * blockDim.x + threadIdx.x;
  long st = (long)gridDim.x * blockDim.x;
  for (; i < n; i += st) p[i] = v;
}

__global__ void k_fill_u32(unsigned* __restrict__ p, unsigned v, long n) {
  long i = (long)blockIdx.x * blockDim.x + threadIdx.x;
  long st = (long)gridDim.x * blockDim.x;
  for (; i < n; i += st) p[i] = v;
}

__global__ void k_init_scal(float* scal) {
  scal[0] = -INFINITY;  // running max M over relation scores
  scal[1] = 0.0f;       // running denom S
  scal[2] = 0.0f;       // current relation score
}

__global__ void k_relu_inplace(float* __restrict__ p, long n) {
  long i = (long)blockIdx.x * blockDim.x + threadIdx.x;
  long st = (long)gridDim.x * blockDim.x;
  for (; i < n; i += st) p[i] = fmaxf(p[i], 0.0f);
}

// ---------------------------------------------------------------------------
// order-preserving float<->uint map for atomicMax-based segment max
// ---------------------------------------------------------------------------
__device__ __forceinline__ unsigned f32_ord(float f) {
  unsigned b = __float_as_uint(f);
  return (b & 0x80000000u) ? ~b : (b | 0x80000000u);
}
__device__ __forceinline__ float ord_f32(unsigned u) {
  unsigned b = (u & 0x80000000u) ? (u ^ 0x80000000u) : ~u;
  return __uint_as_float(b);
}

// ---------------------------------------------------------------------------
// LDS-staged FP32 WMMA GEMM, 8 waves per block, 128x128 output slab.
// A: [M x 128] row-major streamed from global; B: 128x128 staged in LDS.
// Wave w computes rows [blk*128 + w*16, +16) x all 128 cols (8 accumulators),
// so each A fragment feeds 8 V_WMMA_F32_16X16X4_F32 with B frags from ds_load.
// Fragment layout per ISA 7.12.2 (A: lanes 0-15 K{0,1}, 16-31 K{2,3};
// C/D: acc[v] = row v / v+8, col = lane&15).
// ---------------------------------------------------------------------------
__global__ __launch_bounds__(256) void k_wmma_proj_lds(
    const float* __restrict__ A, const float* __restrict__ B,
    const float* __restrict__ bias, float* __restrict__ C, int M) {
  __shared__ float Bs[D128 * D128];
  stage_B_to_lds(B, Bs);

  const int lane = threadIdx.x & 31;
  const int wave = threadIdx.x >> 5;
  const int rc = lane & 15;
  const int kh = (lane >> 4) << 1;  // 0 or 2

  int arow = blockIdx.x * 128 + wave * 16 + rc;
  if (arow >= M) arow = M - 1;  // clamp; stores guarded below
  const float* Ap = A + (size_t)arow * D128;

  v8f acc[8];
  const v8f vz = {0.f, 0.f, 0.f, 0.f, 0.f, 0.f, 0.f, 0.f};
#pragma unroll
  for (int ct = 0; ct < 8; ++ct) acc[ct] = vz;

  for (int k0 = 0; k0 < D128; k0 += 4) {
    v2f a;
    a.x = Ap[k0 + kh];
    a.y = Ap[k0 + kh + 1];
#pragma unroll
    for (int ct = 0; ct < 8; ++ct) {
      v2f b;
      b.x = Bs[(k0 + kh) * D128 + ct * 16 + rc];      // B[k][j] row-major
      b.y = Bs[(k0 + kh + 1) * D128 + ct * 16 + rc];
      acc[ct] = __builtin_amdgcn_wmma_f32_16x16x4_f32(false, a, false, b,
                                                      (short)0, acc[ct], false,
                                                      false);
    }
  }

  const int rtop = blockIdx.x * 128 + wave * 16 + ((lane >> 4) << 3);
#pragma unroll
  for (int ct = 0; ct < 8; ++ct) {
    int col = ct * 16 + rc;
    float bv = bias ? bias[col] : 0.0f;
#pragma unroll
    for (int v = 0; v < 8; ++v) {
      int row = rtop + v;
      if (row < M) C[(size_t)row * D128 + col] = acc[ct][v] + bv;
    }
  }
}

// Same slab structure, B = kw in torch [out,in] layout (so B^T indexing),
// fused epilogue: colsum[j] += sum_rows tanh((A @ kw^T)[n,j] + kb[j]).
__global__ __launch_bounds__(256) void k_wmma_klin_lds(
    const float* __restrict__ A, const float* __restrict__ kw,
    const float* __restrict__ kb, float* __restrict__ colsum, int M) {
  __shared__ float Bs[D128 * D128];
  stage_B_to_lds(kw, Bs);

  const int lane = threadIdx.x & 31;
  const int wave = threadIdx.x >> 5;
  const int rc = lane & 15;
  const int kh = (lane >> 4) << 1;

  int arow = blockIdx.x * 128 + wave * 16 + rc;
  if (arow >= M) arow = M - 1;
  const float* Ap = A + (size_t)arow * D128;

  v8f acc[8];
  const v8f vz = {0.f, 0.f, 0.f, 0.f, 0.f, 0.f, 0.f, 0.f};
#pragma unroll
  for (int ct = 0; ct < 8; ++ct) acc[ct] = vz;

  for (int k0 = 0; k0 < D128; k0 += 4) {
    v2f a;
    a.x = Ap[k0 + kh];
    a.y = Ap[k0 + kh + 1];
#pragma unroll
    for (int ct = 0; ct < 8; ++ct) {
      v2f b;  // Bs[j*128 + k]: consecutive k -> ds_load_b64-friendly
      b.x = Bs[(ct * 16 + rc) * D128 + k0 + kh];
      b.y = Bs[(ct * 16 + rc) * D128 + k0 + kh + 1];
      acc[ct] = __builtin_amdgcn_wmma_f32_16x16x4_f32(false, a, false, b,
                                                      (short)0, acc[ct], false,
                                                      false);
    }
  }

  const int rtop = blockIdx.x * 128 + wave * 16 + ((lane >> 4) << 3);
#pragma unroll
  for (int ct = 0; ct < 8; ++ct) {
    int col = ct * 16 + rc;
    float bv = kb[col];
    float part = 0.0f;
#pragma unroll
    for (int v = 0; v < 8; ++v) {
      int row = rtop + v;
      if (row < M) part += tanhf(acc[ct][v] + bv);
    }
    atomicAdd(&colsum[col], part);
  }
}

// ---------------------------------------------------------------------------
// Small one-wave WMMA GEMM for the [N,128]@[128,16] attention-logit GEMM.
// B^T is the packed 16x128 [asrc; adst] tile (8KB, stays hot in cache).
// ---------------------------------------------------------------------------
__global__ __launch_bounds__(32) void k_wmma_gemm_bt16(
    const float* __restrict__ A, const float* __restrict__ Bt,
    float* __restrict__ C, int M) {
  const int lane = threadIdx.x;
  const int rowBase = blockIdx.x * 16;
  const int rc = lane & 15;
  const int kh = (lane >> 4) << 1;

  int arow = rowBase + rc;
  if (arow >= M) arow = M - 1;
  const float* Ap = A + (size_t)arow * D128;
  const float* Bp = Bt + (size_t)rc * D128;

  v8f acc = {0.f, 0.f, 0.f, 0.f, 0.f, 0.f, 0.f, 0.f};
  for (int k0 = 0; k0 < D128; k0 += 4) {
    v2f a, b;
    a.x = Ap[k0 + kh];
    a.y = Ap[k0 + kh + 1];
    b.x = Bp[k0 + kh];
    b.y = Bp[k0 + kh + 1];
    acc = __builtin_amdgcn_wmma_f32_16x16x4_f32(false, a, false, b, (short)0,
                                                acc, false, false);
  }

  const int rtop = rowBase + ((lane >> 4) << 3);
#pragma unroll
  for (int v = 0; v < 8; ++v) {
    int row = rtop + v;
    if (row < M) C[(size_t)row * 16 + rc] = acc[v];
  }
}

// pack [asrc(8x128); adst(8x128)] into one 16x128 B^T tile for the att GEMM
__global__ void k_build_attB(const float* __restrict__ asrc,
                             const float* __restrict__ adst,
                             float* __restrict__ attB) {
  int i = blockIdx.x * blockDim.x + threadIdx.x;
  if (i >= 16 * D128) return;
  int j = i >> 7, k = i & 127;
  attB[i] = (j < 8) ? asrc[j * D128 + k] : adst[(j - 8) * D128 + k];
}

// ---------------------------------------------------------------------------
// edge-phase kernels (segment softmax + weighted scatter aggregation)
// acat: [N x 16]  cols 0..7 = a_src[n,r], cols 8..15 = a_dst[n,r]
// ---------------------------------------------------------------------------
__device__ __forceinline__ float edge_logit(const float* acat, int src, int dst,
                                            int r) {
  float raw = acat[(size_t)src * 16 + r] + acat[(size_t)dst * 16 + 8 + r];
  return raw >= 0.0f ? raw : NSLOPE * raw;
}

__global__ void k_edge_max(const int* __restrict__ ei,
                           const float* __restrict__ acat,
                           unsigned* __restrict__ menc, int E, int r) {
  int e = blockIdx.x * blockDim.x + threadIdx.x;
  if (e >= E) return;
  int src = ei[(size_t)(2 * r) * E + e];
  int dst = ei[(size_t)(2 * r + 1) * E + e];
  atomicMax(&menc[dst], f32_ord(edge_logit(acat, src, dst, r)));
}

__global__ void k_edge_exp(const int* __restrict__ ei,
                           const float* __restrict__ acat,
                           const unsigned* __restrict__ menc,
                           float* __restrict__ ssum, float* __restrict__ alpha,
                           const float* __restrict__ h, int E, int r) {
  int e = blockIdx.x * blockDim.x + threadIdx.x;
  if (e >= E) return;
  int src = ei[(size_t)(2 * r) * E + e];
  int dst = ei[(size_t)(2 * r + 1) * E + e];
  float m = ord_f32(menc[dst]);
  float ex = expf(edge_logit(acat, src, dst, r) - m);
  alpha[e] = ex;
  atomicAdd(&ssum[dst], ex);
  // warm L2 with the h[src] row the aggregation pass will gather
  __builtin_prefetch(&h[(size_t)src * D128], 0, 0);
}

__global__ __launch_bounds__(128) void k_edge_agg(
    const int* __restrict__ ei, const float* __restrict__ h,
    const float* __restrict__ alpha, const float* __restrict__ ssum,
    float* __restrict__ tmp, int E, int r) {
  int e = blockIdx.x;
  int src = ei[(size_t)(2 * r) * E + e];
  int dst = ei[(size_t)(2 * r + 1) * E + e];
  float w = alpha[e] / (ssum[dst] + 1e-16f);
  int k = threadIdx.x;
  atomicAdd(&tmp[(size_t)dst * D128 + k], w * h[(size_t)src * D128 + k]);
}

// ---------------------------------------------------------------------------
// semantic attention (online softmax over relations)
// ---------------------------------------------------------------------------
__global__ void k_score_fin(const float* __restrict__ colsum,
                            const float* __restrict__ q, float* scal,
                            float invN) {
  __shared__ float red[D128];
  int t = threadIdx.x;
  red[t] = colsum[t] * q[t];
  __syncthreads();
  for (int s = 64; s > 0; s >>= 1) {
    if (t < s) red[t] += red[t + s];
    __syncthreads();
  }
  if (t == 0) scal[2] = red[0] * invN;
}

__global__ void k_v_update(float* __restrict__ V, const float* __restrict__ tmp,
                           const float* __restrict__ scal, long n) {
  float m_old = scal[0], sc = scal[2];
  float mn = fmaxf(m_old, sc);
  float f0 = expf(m_old - mn);  // 0 on first relation (m_old = -inf)
  float f1 = expf(sc - mn);
  long i = (long)blockIdx.x * blockDim.x + threadIdx.x;
  long st = (long)gridDim.x * blockDim.x;
  for (; i < n; i += st) V[i] = V[i] * f0 + f1 * tmp[i];
}

__global__ void k_ms_update(float* scal) {
  float m_old = scal[0], sc = scal[2];
  float mn = fmaxf(m_old, sc);
  scal[1] = scal[1] * expf(m_old - mn) + expf(sc - mn);
  scal[0] = mn;
}

__global__ void k_layer_final(const float* __restrict__ V,
                              const float* __restrict__ scal,
                              float* __restrict__ xout, long n) {
  float invS = 1.0f / scal[1];
  long i = (long)blockIdx.x * blockDim.x + threadIdx.x;
  long st = (long)gridDim.x * blockDim.x;
  for (; i < n; i += st) xout[i] = fmaxf(V[i] * invS, 0.0f);
}

// ---------------------------------------------------------------------------
// link head: out[l] = sum_k h[s,k]*h[d,k]*(pw[k,0]+pw[k,1]) + (pb0+pb1)
// one wave32 per label pair, float4 loads, shfl_xor reduce
// ---------------------------------------------------------------------------
__global__ __launch_bounds__(256) void k_head(
    const float* __restrict__ h2, const int* __restrict__ eli,
    const float* __restrict__ pw, const float* __restrict__ pb,
    float* __restrict__ out, int L) {
  int warp = threadIdx.x >> 5, lane = threadIdx.x & 31;
  int l = blockIdx.x * 8 + warp;
  if (l >= L) return;
  int s = eli[l];
  int d = eli[(size_t)L + l];
  const float4* hs = (const float4*)(h2 + (size_t)s * D128);
  const float4* hd = (const float4*)(h2 + (size_t)d * D128);
  float4 a = hs[lane], b = hd[lane];
  int k = lane * 4;
  float p0 = pw[2 * k + 0] + pw[2 * k + 1];
  float p1 = pw[2 * k + 2] + pw[2 * k + 3];
  float p2 = pw[2 * k + 4] + pw[2 * k + 5];
  float p3 = pw[2 * k + 6] + pw[2 * k + 7];
  float part = a.x * b.x * p0 + a.y * b.y * p1 + a.z * b.z * p2 + a.w * b.w * p3;
  for (int off = 16; off > 0; off >>= 1) part += __shfl_xor(part, off, 32);
  if (lane == 0) out[l] = part + pb[0] + pb[1];
}

// ---------------------------------------------------------------------------
// host orchestration
// ---------------------------------------------------------------------------
static void run_layer(const float* xin, float* xout, const float* w,
                      const float* b, const float* asrc, const float* adst,
                      const float* q, const float* kw, const float* kb,
                      const int* ei, int N, int E, int R, float* h, float* V,
                      float* tmp, float* acat, unsigned* menc, float* ssum,
                      float* alpha, float* attB, float* colsum, float* scal,
                      hipStream_t stream) {
  const long ND = (long)N * D128;
  const int slabs = (N + 127) / 128;
  const int rtiles = (N + 15) / 16;

  // h = xin @ W + b   (LDS-staged slab GEMM)
  k_wmma_proj_lds<<<slabs, 256, 0, stream>>>(xin, w, b, h, N);
  // a_src | a_dst logits via one [N,128]@[128,16] WMMA GEMM
  k_build_attB<<<(16 * D128 + 255) / 256, 256, 0, stream>>>(asrc, adst, attB);
  k_wmma_gemm_bt16<<<rtiles, 32, 0, stream>>>(h, attB, acat, N);

  k_init_scal<<<1, 1, 0, stream>>>(scal);
  k_fill_f32<<<1024, 256, 0, stream>>>(V, 0.0f, ND);

  int eb = (E + 255) / 256;
  for (int r = 0; r < R; ++r) {
    k_fill_u32<<<256, 256, 0, stream>>>(menc, 0u, N);
    k_fill_f32<<<256, 256, 0, stream>>>(ssum, 0.0f, N);
    k_fill_f32<<<1024, 256, 0, stream>>>(tmp, 0.0f, ND);
    // segment softmax over in-edges of each dst, then weighted aggregation
    k_edge_max<<<eb, 256, 0, stream>>>(ei, acat, menc, E, r);
    k_edge_exp<<<eb, 256, 0, stream>>>(ei, acat, menc, ssum, alpha, h, E, r);
    k_edge_agg<<<E, 128, 0, stream>>>(ei, h, alpha, ssum, tmp, E, r);
    k_relu_inplace<<<1024, 256, 0, stream>>>(tmp, ND);
    // score_r = q . mean_n tanh(out_r @ kw^T + kb)   (fused WMMA + reduce)
    k_fill_f32<<<1, 128, 0, stream>>>(colsum, 0.0f, D128);
    k_wmma_klin_lds<<<slabs, 256, 0, stream>>>(tmp, kw, kb, colsum, N);
    k_score_fin<<<1, 128, 0, stream>>>(colsum, q, scal, 1.0f / (float)N);
    // online softmax accumulate: V = V*exp(M-M') + exp(score-M')*out_r
    k_v_update<<<1024, 256, 0, stream>>>(V, tmp, scal, ND);
    k_ms_update<<<1, 1, 0, stream>>>(scal);
  }
  // xout = relu(V / S)
  k_layer_final<<<1024, 256, 0, stream>>>(V, scal, xout, ND);
}

extern "C" void kernel_launch(void* const* d_in, const int* in_sizes, int n_in,
                              void* d_out, int out_size, void* d_ws,
                              size_t ws_size, hipStream_t stream) {
  const int R = 8;
  const int N = in_sizes[0] / D128;
  const int E = in_sizes[1] / (R * 2);
  const int L = in_sizes[2] / 2;

  const float* x = (const float*)d_in[0];
  const int* ei = (const int*)d_in[1];
  const int* eli = (const int*)d_in[2];
  const float* w1 = (const float*)d_in[3];
  const float* b1 = (const float*)d_in[4];
  const float* asrc1 = (const float*)d_in[5];
  const float* adst1 = (const float*)d_in[6];
  const float* q1 = (const float*)d_in[7];
  const float* kw1 = (const float*)d_in[8];
  const float* kb1 = (const float*)d_in[9];
  const float* w2 = (const float*)d_in[10];
  const float* b2 = (const float*)d_in[11];
  const float* asrc2 = (const float*)d_in[12];
  const float* adst2 = (const float*)d_in[13];
  const float* q2 = (const float*)d_in[14];
  const float* kw2 = (const float*)d_in[15];
  const float* kb2 = (const float*)d_in[16];
  const float* postw = (const float*)d_in[17];
  const float* postb = (const float*)d_in[18];

  // workspace carve (~213 MB total for N=100k, E=200k)
  char* p = (char*)d_ws;
  auto carve = [&](size_t bytes) -> char* {
    char* r = p;
    p += (bytes + 255) & ~(size_t)255;
    return r;
  };
  float* h = (float*)carve((size_t)N * D128 * 4);
  float* x1 = (float*)carve((size_t)N * D128 * 4);
  float* V = (float*)carve((size_t)N * D128 * 4);
  float* tmp = (float*)carve((size_t)N * D128 * 4);
  float* acat = (float*)carve((size_t)N * 16 * 4);
  unsigned* menc = (unsigned*)carve((size_t)N * 4);
  float* ssum = (float*)carve((size_t)N * 4);
  float* alpha = (float*)carve((size_t)E * 4);
  float* attB = (float*)carve((size_t)16 * D128 * 4);
  float* colsum = (float*)carve((size_t)D128 * 4);
  float* scal = (float*)carve(16);

  // layer 1: x -> x1 ;  layer 2: x1 -> tmp (tmp free after its internal loop)
  run_layer(x, x1, w1, b1, asrc1, adst1, q1, kw1, kb1, ei, N, E, R, h, V, tmp,
            acat, menc, ssum, alpha, attB, colsum, scal, stream);
  run_layer(x1, tmp, w2, b2, asrc2, adst2, q2, kw2, kb2, ei, N, E, R, h, V, tmp,
            acat, menc, ssum, alpha, attB, colsum, scal, stream);

  k_head<<<(L + 7) / 8, 256, 0, stream>>>(tmp, eli, postw, postb,
                                          (float*)d_out, L);
}